// AnnAttention_70480413327911
// MI455X (gfx1250) — compile-verified
//
#include <hip/hip_runtime.h>
#include <math.h>

// ---------------- problem constants ----------------
#define Bc     4
#define KVHc   8
#define HPKc   4
#define Dc     128
#define SEQKV  4096
#define SEQc   4097
#define LOCALK 128
#define SINKc  4
#define RANKc  32
#define KK     516          // min(K+SINK, SEQ)
#define NPAIR  (Bc*KVHc)    // 32

// ---------------- workspace layout (dword units, per (b,kvh) pair) --------
#define WS_CHAN   0
#define WS_SCALE  32
#define WS_QPROJ  36
#define WS_KVW    164
#define WS_SCORE  168
#define WS_TSUM   (WS_SCORE + 4*SEQc)          // 16556
#define WS_SEL    (WS_TSUM + SEQKV)            // 20652
#define WS_SPEC   (WS_SEL + 520)               // 21172
#define WS_PAIR   (WS_SPEC + 4096)             // 25268 dwords / pair

typedef __attribute__((ext_vector_type(2))) float v2f;
typedef __attribute__((ext_vector_type(8))) float v8f;

// ---------------- TDM (Tensor Data Mover) gather-mode support -------------
#if defined(__gfx1250__) && __has_builtin(__builtin_amdgcn_tensor_load_to_lds) && __has_builtin(__builtin_amdgcn_s_wait_tensorcnt)
#define HAVE_TDM 1
#else
#define HAVE_TDM 0
#endif

#if HAVE_TDM
typedef __attribute__((ext_vector_type(4))) unsigned int u32x4;
typedef __attribute__((ext_vector_type(8))) int          i32x8;
typedef __attribute__((ext_vector_type(4))) int          i32x4;

// Gather up to 16 rows (16-bit indices) of `tile_w_dw` dwords each from
// gaddr + idx*row_stride_dw*4 into LDS at lds_addr, with LDS row padding
// (pad codes per D# group1: interval code i -> 2^(i+1) dwords, amount code
// a -> a+1 dwords).  D# bitfields per CDNA5 ISA 8.3-8.7.
// 6-arg builtin arity (clang-23 / therock headers): groups 0..3 + zero VADDR4
// group + cache-policy immediate last.
__device__ __forceinline__ void tdm_gather(const void* gptr, unsigned lds_addr,
                                           const int* idx, int nidx,
                                           int tile_w_dw, long long row_stride_dw,
                                           int pad_itv, int pad_amt) {
  unsigned long long ga = (unsigned long long)gptr;
  u32x4 g0 = { 0x80000001u,                                   // count=1, gather_mode=1, 16b idx
               lds_addr,
               (unsigned)(ga & 0xFFFFFFFFu),
               (unsigned)((ga >> 32) & 0x01FFFFFFu) | (2u << 30) }; // addr[56:32], type=2
  unsigned td0 = (unsigned)tile_w_dw;                         // tensor_dim0 (elements)
  unsigned td1 = (unsigned)SEQKV;                             // tensor_dim1 (rows)
  int dw0 = (2 << 16) | (1 << 20) | (pad_itv << 22) | (pad_amt << 25); // 4B elems, pad_enable
  i32x8 g1 = { dw0,
               (int)(td0 << 16),                              // [47:32]=bar_addr=0 | td0 lo16
               (int)((td0 >> 16) | (td1 << 16)),              // td0 hi16 | td1 lo16
               (int)((td1 >> 16) | ((unsigned)tile_w_dw << 16)), // td1 hi16 | tile_dim0
               (int)(nidx & 0xFFFF),                          // tile_dim1 = #valid indices
               (int)(unsigned)(row_stride_dw & 0xFFFFFFFFll), // dim0_stride lo32
               (int)(unsigned)((row_stride_dw >> 32) & 0xFFFFll),
               0 };
  int p[16];
  for (int i = 0; i < 16; i++) p[i] = (i < nidx) ? (idx[i] & 0xFFFF) : 0;
  i32x4 g2 = { p[0] | (p[1] << 16),  p[2] | (p[3] << 16),
               p[4] | (p[5] << 16),  p[6] | (p[7] << 16) };
  i32x4 g3 = { p[8] | (p[9] << 16),  p[10] | (p[11] << 16),
               p[12] | (p[13] << 16), p[14] | (p[15] << 16) };
  i32x8 g4 = { 0, 0, 0, 0, 0, 0, 0, 0 };                      // unused VADDR4 group
  __builtin_amdgcn_tensor_load_to_lds(g0, g1, g2, g3, g4, 0);
}
#endif

// ================= K1: channel top-32 selection ==========================
__global__ void __launch_bounds__(128) k_chan(const float* __restrict__ q,
                                              float* __restrict__ ws) {
  int pair = blockIdx.x;                 // b*KVH + kvh
  int tid  = threadIdx.x;                // = d (0..127)
  __shared__ float qs[HPKc][Dc];
  __shared__ float absq[Dc];
  __shared__ float red[Dc];
  __shared__ int   redi[Dc];
  __shared__ int   chosen[RANKc];
  const float* qb = q + (size_t)pair * HPKc * Dc;
  float s = 0.f;
  for (int h = 0; h < HPKc; h++) { float v = qb[h*Dc + tid]; qs[h][tid] = v; s += fabsf(v); }
  absq[tid] = s;
  __syncthreads();
  for (int r = 0; r < RANKc; r++) {
    red[tid] = absq[tid]; redi[tid] = tid;
    __syncthreads();
    for (int off = 64; off > 0; off >>= 1) {
      if (tid < off) {
        float a = red[tid], b2 = red[tid+off];
        int ia = redi[tid], ib = redi[tid+off];
        if (b2 > a || (b2 == a && ib < ia)) { red[tid] = b2; redi[tid] = ib; }
      }
      __syncthreads();
    }
    if (tid == 0) { chosen[r] = redi[0]; absq[redi[0]] = -1.0f; }
    __syncthreads();
  }
  size_t base = (size_t)pair * WS_PAIR;
  if (tid < RANKc) ((int*)ws)[base + WS_CHAN + tid] = chosen[tid];
  { int h = tid >> 5, r = tid & 31; ws[base + WS_QPROJ + tid] = qs[h][chosen[r]]; }
  if (tid < HPKc) {
    float sf = 0.f, sp = 0.f;
    for (int d = 0; d < Dc; d++)    sf += fabsf(qs[tid][d]);
    for (int r = 0; r < RANKc; r++) sp += fabsf(qs[tid][chosen[r]]);
    ws[base + WS_SCALE + tid] = sqrtf(sp / sf * (float)Dc);
  }
}

// ============ K2: logmask -> local/sink "special" flags (suffix scan) =====
__global__ void __launch_bounds__(256) k_flags(const float* __restrict__ lm,
                                               float* __restrict__ ws) {
  int row = blockIdx.x;                  // pair*4 + h
  int pair = row >> 2, h = row & 3;
  int tid = threadIdx.x;
  const float* lrow = lm + (size_t)row * SEQc;
  __shared__ int cnt[256];
  int neg[16]; int c = 0;
  int ebase = tid * 16;
  for (int i = 0; i < 16; i++) { float v = lrow[ebase + i]; int n = (v == -INFINITY); neg[i] = n; c += !n; }
  cnt[tid] = c;
  __syncthreads();
  for (int off = 1; off < 256; off <<= 1) {          // inclusive scan
    int add = (tid >= off) ? cnt[tid - off] : 0;
    __syncthreads();
    cnt[tid] += add;
    __syncthreads();
  }
  int total  = cnt[255];
  int prefix = cnt[tid] - c;
  unsigned char* spec = ((unsigned char*)ws) + ((size_t)pair * WS_PAIR + WS_SPEC) * 4 + (size_t)h * SEQKV;
  int run = prefix;
  for (int i = 0; i < 16; i++) {
    int nonneg = !neg[i];
    int suffix = total - run;                         // non-neg count in [j, end]
    spec[ebase + i] = (unsigned char)(nonneg && (suffix <= LOCALK || suffix >= (SEQc - SINKc)));
    run += nonneg;
  }
}

// ================= K3: sparse score via WMMA f32 16x16x4 =================
__global__ void __launch_bounds__(256) k_score(const float* __restrict__ skey,
                                               const float* __restrict__ keycur,
                                               const float* __restrict__ lm,
                                               float* __restrict__ ws) {
  int pair = blockIdx.x;
  int tid = threadIdx.x, lane = tid & 31, wv = tid >> 5;   // 8 waves
  size_t base = (size_t)pair * WS_PAIR;
  __shared__ float aq[HPKc][RANKc];
  __shared__ float lk[RANKc][264];        // pad 264: 2*264 % 64 == 16 -> conflict-free B reads
  __shared__ int   ch[RANKc];
  __shared__ float scinv[HPKc];
  if (tid < 128) aq[tid >> 5][tid & 31] = ws[base + WS_QPROJ + tid];
  if (tid < RANKc) ch[tid] = ((int*)ws)[base + WS_CHAN + tid];
  if (tid < HPKc)  scinv[tid] = 1.0f / ws[base + WS_SCALE + tid];
  __syncthreads();

  int m = lane & 15;
  int koff = (lane < 16) ? 0 : 2;
  v2f afrag[8];                            // A is 16x32 (4 live rows), 8 K-steps of 4
  for (int s = 0; s < 8; s++) {
    float ax = aq[m & 3][4*s + koff];
    float ay = aq[m & 3][4*s + koff + 1];
    afrag[s].x = (m < 4) ? ax : 0.f;
    afrag[s].y = (m < 4) ? ay : 0.f;
  }

  const float* srow_base = skey + (size_t)pair * Dc * SEQKV;
  const float* lmrow = lm + (size_t)pair * HPKc * SEQc;
  const unsigned char* spec = ((const unsigned char*)ws) + (base + WS_SPEC) * 4;
  const float FMAXV = 3.402823466e38f;

  for (int chunk = 0; chunk < 16; chunk++) {
    int j0 = chunk * 256;
#if HAVE_TDM
    // TDM gather: 2 descriptors x 16 rows x 256 dwords, padded to 264-dw rows
    if (tid < 32) {
      unsigned lbase = (unsigned)(unsigned long long)(void*)&lk[0][0];
      tdm_gather(srow_base + j0, lbase,                 &ch[0],  16, 256, SEQKV, 7, 7);
      tdm_gather(srow_base + j0, lbase + 16u*264u*4u,   &ch[16], 16, 256, SEQKV, 7, 7);
      __builtin_amdgcn_s_wait_tensorcnt(0);
    }
#else
    for (int e = tid; e < RANKc * 256; e += 256) {
      int c = e >> 8, j = e & 255;
      lk[c][j] = srow_base[(size_t)ch[c] * SEQKV + j0 + j];
    }
#endif
    if (chunk < 15) {                     // global_prefetch_b8 next chunk
      int r = tid >> 3, seg = tid & 7;
      __builtin_prefetch(srow_base + (size_t)ch[r] * SEQKV + j0 + 256 + seg * 32, 0, 0);
    }
    __syncthreads();
    for (int tt = 0; tt < 2; tt++) {
      int n0 = (wv + tt * 8) * 16;
      v8f acc = {};
      for (int s = 0; s < 8; s++) {
        v2f bfr;
        bfr.x = lk[4*s + koff][n0 + m];
        bfr.y = lk[4*s + koff + 1][n0 + m];
        acc = __builtin_amdgcn_wmma_f32_16x16x4_f32(false, afrag[s], false, bfr,
                                                    (short)0, acc, false, false);
      }
      if (lane < 16) {                    // heads are C rows 0..3 -> lanes 0-15
        int j = j0 + n0 + m;
        float tsum = 0.f;
        for (int h = 0; h < HPKc; h++) {
          float scv = acc[h] * scinv[h] + lmrow[(size_t)h * SEQc + j];
          ws[base + WS_SCORE + (size_t)h * SEQc + j] = scv;
          tsum += spec[(size_t)h * SEQKV + j] ? FMAXV : scv;
        }
        ws[base + WS_TSUM + j] = tsum;
      }
    }
    __syncthreads();
  }
  // current key (j = 4096)
  if (tid < HPKc) {
    const float* kc = keycur + (size_t)pair * Dc;
    float d = 0.f;
    for (int r = 0; r < RANKc; r++) d += aq[tid][r] * kc[ch[r]];
    ws[base + WS_SCORE + (size_t)tid * SEQc + SEQKV] = d * scinv[tid] + lmrow[(size_t)tid * SEQc + SEQKV];
  }
}

// ====== K4: exact top-516 (descending, stable) via bitonic sort in LDS ====
__global__ void __launch_bounds__(1024) k_topk(float* __restrict__ ws) {
  int pair = blockIdx.x, tid = threadIdx.x;
  size_t base = (size_t)pair * WS_PAIR;
  __shared__ unsigned long long keys[SEQKV];
  for (int i = tid; i < SEQKV; i += 1024) {
    unsigned u = __float_as_uint(ws[base + WS_TSUM + i]);
    u = (u & 0x80000000u) ? ~u : (u | 0x80000000u);   // order-preserving transform
    keys[i] = ((unsigned long long)u << 32) | (unsigned)(~i); // ties -> lower index first
  }
  __syncthreads();
  for (int k = 2; k <= SEQKV; k <<= 1) {
    for (int j = k >> 1; j > 0; j >>= 1) {
      for (int i = tid; i < SEQKV; i += 1024) {
        int ixj = i ^ j;
        if (ixj > i) {
          unsigned long long a = keys[i], b2 = keys[ixj];
          bool desc = ((i & k) == 0);
          if (desc ? (a < b2) : (a > b2)) { keys[i] = b2; keys[ixj] = a; }
        }
      }
      __syncthreads();
    }
  }
  for (int t = tid; t < KK; t += 1024)
    ((int*)ws)[base + WS_SEL + t] = (int)(~(unsigned)(keys[t] & 0xFFFFFFFFu));
}

// ============ K5: softmax stats over 4097 + kv_weight ====================
__global__ void __launch_bounds__(256) k_kvw(float* __restrict__ ws) {
  int row = blockIdx.x;                   // pair*4 + h
  int pair = row >> 2, h = row & 3;
  int tid = threadIdx.x;
  size_t base = (size_t)pair * WS_PAIR;
  const float* s = ws + base + WS_SCORE + (size_t)h * SEQc;
  __shared__ float red[256];
  float mx = -INFINITY;
  for (int i = tid; i < SEQc; i += 256) mx = fmaxf(mx, s[i]);
  red[tid] = mx; __syncthreads();
  for (int off = 128; off > 0; off >>= 1) { if (tid < off) red[tid] = fmaxf(red[tid], red[tid+off]); __syncthreads(); }
  mx = red[0]; __syncthreads();
  float sm = 0.f;
  for (int i = tid; i < SEQc; i += 256) sm += expf(s[i] - mx);
  red[tid] = sm; __syncthreads();
  for (int off = 128; off > 0; off >>= 1) { if (tid < off) red[tid] += red[tid+off]; __syncthreads(); }
  float denom = red[0]; __syncthreads();
  const int* sel = ((const int*)ws) + base + WS_SEL;
  float ss = 0.f;
  for (int t = tid; t < KK; t += 256) ss += expf(s[sel[t]] - mx);
  red[tid] = ss; __syncthreads();
  for (int off = 128; off > 0; off >>= 1) { if (tid < off) red[tid] += red[tid+off]; __syncthreads(); }
  if (tid == 0)
    ws[base + WS_KVW + h] = (red[0] + expf(s[SEQKV] - mx)) / denom;
}

// ===== K6: gathered attention: WMMA q.Kg^T, softmax*kv_w, w.Vg + mean_v ===
__global__ void __launch_bounds__(256) k_final(const float* __restrict__ q,
    const float* __restrict__ keycur, const float* __restrict__ valcur,
    const float* __restrict__ lm, const float* __restrict__ key_cpu,
    const float* __restrict__ val_cpu, const float* __restrict__ mean_v,
    float* __restrict__ ws, float* __restrict__ out) {
  int pair = blockIdx.x;
  int tid = threadIdx.x, lane = tid & 31, wv = tid >> 5;
  size_t base = (size_t)pair * WS_PAIR;
  __shared__ float qlds[HPKc][Dc];
  __shared__ float rows[64][132];          // pad 132: conflict-free [n][c] B reads
  __shared__ float s2[HPKc][544];
  __shared__ float wts[HPKc][544];
  __shared__ float red[256];
  __shared__ float kvv[HPKc];
  __shared__ int   selS[520];
  for (int e = tid; e < HPKc * Dc; e += 256)
    qlds[e >> 7][e & 127] = q[(size_t)(pair * HPKc + (e >> 7)) * Dc + (e & 127)];
  for (int e = tid; e < KK; e += 256) selS[e] = ((const int*)ws)[base + WS_SEL + e];
  if (tid < HPKc) kvv[tid] = ws[base + WS_KVW + tid];
  __syncthreads();

  int m = lane & 15;
  int koff = (lane < 16) ? 0 : 2;
  const float* kcp = key_cpu + (size_t)pair * SEQKV * Dc;
  const float* vcp = val_cpu + (size_t)pair * SEQKV * Dc;
  const float RSQRT_D = 0.08838834764831845f;       // 128^-0.5

  // ---- stage 1: scores2[h][t] = q.key_g^T / sqrt(D) + lm_g ----
  for (int cc = 0; cc < 9; cc++) {
#if HAVE_TDM
    if (tid < 32) {
      unsigned lbase = (unsigned)(unsigned long long)(void*)&rows[0][0];
      for (int gi = 0; gi < 4; gi++) {
        int t0 = cc * 64 + gi * 16;
        int n = KK - t0; if (n > 16) n = 16;
        if (n > 0) tdm_gather(kcp, lbase + (unsigned)gi*16u*132u*4u, &selS[t0], n, Dc, Dc, 6, 3);
      }
      __builtin_amdgcn_s_wait_tensorcnt(0);
    }
    if (cc == 8 && tid < Dc) rows[KK - 512][tid] = keycur[(size_t)pair * Dc + tid];
#else
    for (int e = tid; e < 64 * Dc; e += 256) {
      int r = e >> 7, d = e & 127;
      int t = cc * 64 + r;
      float v = 0.f;
      if (t < KK)       v = kcp[(size_t)selS[t] * Dc + d];
      else if (t == KK) v = keycur[(size_t)pair * Dc + d];
      rows[r][d] = v;
    }
#endif
    __syncthreads();
    if (wv < 4) {
      int n0 = wv * 16;
      v8f acc = {};
      for (int s = 0; s < 32; s++) {                 // K = 128 in steps of 4
        int c0 = 4 * s + koff;
        v2f a, b2;
        float ax = qlds[m & 3][c0], ay = qlds[m & 3][c0 + 1];
        a.x = (m < 4) ? ax : 0.f;
        a.y = (m < 4) ? ay : 0.f;
        b2.x = rows[n0 + m][c0];
        b2.y = rows[n0 + m][c0 + 1];
        acc = __builtin_amdgcn_wmma_f32_16x16x4_f32(false, a, false, b2,
                                                    (short)0, acc, false, false);
      }
      if (lane < 16) {
        int t = cc * 64 + n0 + m;
        if (t <= KK) {
          int li = (t < KK) ? selS[t] : (SEQc - 1);
          for (int h = 0; h < HPKc; h++)
            s2[h][t] = acc[h] * RSQRT_D + lm[(size_t)(pair * HPKc + h) * SEQc + li];
        }
      }
    }
    __syncthreads();
  }

  // ---- softmax per head, scaled by kv_weight; emit `weights` output ----
  for (int h = 0; h < HPKc; h++) {
    float mx = -INFINITY;
    for (int t = tid; t < KK + 1; t += 256) mx = fmaxf(mx, s2[h][t]);
    red[tid] = mx; __syncthreads();
    for (int off = 128; off > 0; off >>= 1) { if (tid < off) red[tid] = fmaxf(red[tid], red[tid+off]); __syncthreads(); }
    mx = red[0]; __syncthreads();
    float sm = 0.f;
    for (int t = tid; t < KK + 1; t += 256) sm += expf(s2[h][t] - mx);
    red[tid] = sm; __syncthreads();
    for (int off = 128; off > 0; off >>= 1) { if (tid < off) red[tid] += red[tid+off]; __syncthreads(); }
    float scale = kvv[h] / red[0]; __syncthreads();
    for (int t = tid; t < KK + 1; t += 256) {
      float w = expf(s2[h][t] - mx) * scale;
      wts[h][t] = w;
      out[(size_t)(Bc*KVHc*HPKc*Dc) + (size_t)(pair * HPKc + h) * (KK + 1) + t] = w;
    }
    __syncthreads();
  }

  // ---- stage 2: out = weights @ val_g + (1-kv)*mean_v_new ----
  float acc0 = 0.f, acc1 = 0.f;
  int h0 = tid >> 7, d0 = tid & 127;       // heads 0,1
  int h1 = h0 + 2;                         // heads 2,3
  for (int cc = 0; cc < 9; cc++) {
#if HAVE_TDM
    if (tid < 32) {
      unsigned lbase = (unsigned)(unsigned long long)(void*)&rows[0][0];
      for (int gi = 0; gi < 4; gi++) {
        int t0 = cc * 64 + gi * 16;
        int n = KK - t0; if (n > 16) n = 16;
        if (n > 0) tdm_gather(vcp, lbase + (unsigned)gi*16u*132u*4u, &selS[t0], n, Dc, Dc, 6, 3);
      }
      __builtin_amdgcn_s_wait_tensorcnt(0);
    }
    if (cc == 8 && tid < Dc) rows[KK - 512][tid] = valcur[(size_t)pair * Dc + tid];
#else
    for (int e = tid; e < 64 * Dc; e += 256) {
      int r = e >> 7, d = e & 127;
      int t = cc * 64 + r;
      float v = 0.f;
      if (t < KK)       v = vcp[(size_t)selS[t] * Dc + d];
      else if (t == KK) v = valcur[(size_t)pair * Dc + d];
      rows[r][d] = v;
    }
#endif
    __syncthreads();
    int rv = KK + 1 - cc * 64; if (rv > 64) rv = 64;
    for (int r = 0; r < rv; r++) {
      acc0 += wts[h0][cc * 64 + r] * rows[r][d0];
      acc1 += wts[h1][cc * 64 + r] * rows[r][d0];
    }
    __syncthreads();
  }
  float mvn = (mean_v[(size_t)pair * Dc + d0] * (float)SEQKV +
               valcur[(size_t)pair * Dc + d0]) / (float)SEQc;
  out[(size_t)(pair * HPKc + h0) * Dc + d0] = acc0 + (1.f - kvv[h0]) * mvn;
  out[(size_t)(pair * HPKc + h1) * Dc + d0] = acc1 + (1.f - kvv[h1]) * mvn;
}

// ========================= host launcher =================================
extern "C" void kernel_launch(void* const* d_in, const int* in_sizes, int n_in,
                              void* d_out, int out_size, void* d_ws, size_t ws_size,
                              hipStream_t stream) {
  (void)in_sizes; (void)n_in; (void)out_size; (void)ws_size;
  const float* query   = (const float*)d_in[0];
  const float* keycur  = (const float*)d_in[1];
  const float* valcur  = (const float*)d_in[2];
  const float* logmask = (const float*)d_in[3];
  const float* key_cpu = (const float*)d_in[4];
  const float* val_cpu = (const float*)d_in[5];
  const float* s_key   = (const float*)d_in[6];
  const float* mean_v  = (const float*)d_in[7];
  float* out = (float*)d_out;
  float* ws  = (float*)d_ws;

  k_chan <<<NPAIR,        128,  0, stream>>>(query, ws);
  k_flags<<<NPAIR*HPKc,   256,  0, stream>>>(logmask, ws);
  k_score<<<NPAIR,        256,  0, stream>>>(s_key, keycur, logmask, ws);
  k_topk <<<NPAIR,        1024, 0, stream>>>(ws);
  k_kvw  <<<NPAIR*HPKc,   256,  0, stream>>>(ws);
  k_final<<<NPAIR,        256,  0, stream>>>(query, keycur, valcur, logmask,
                                             key_cpu, val_cpu, mean_v, ws, out);
}